// CartesianRealSHT_3770981286258
// MI455X (gfx1250) — compile-verified
//
#include <hip/hip_runtime.h>
#include <hip/hip_bf16.h>

typedef float v2f __attribute__((ext_vector_type(2)));
typedef float v4f __attribute__((ext_vector_type(4)));
typedef float v8f __attribute__((ext_vector_type(8)));

#define NLAT 361
#define NLON 720
#define MMAX 361
#define LMAX 361
#define NCH  32
#define MPAD 368                  // MMAX padded to multiple of 16
#define NROW (NCH * NLAT)         // 11552 = 722 * 16 exactly
#define NTILER (NROW / 16)        // 722
#define NTILEM (MPAD / 16)        // 23
#define NSTEP1 (NLON / 4)         // 180 k-steps in stage 1
#define KSTEP2 91                 // ceil(361/4) k-steps in stage 2

#define TWO_PI 6.28318530717958647692f

// ---------------------------------------------------------------------------
// Workspace layout (floats):
//   PF : packed DFT basis fragments  [mtile(23)][nstep(180)][lane(32)][4]
//        lane<16: (cos(na,m), cos(na+1,m), -sin(na,m), -sin(na+1,m)) * s
//        where m = mtile*16 + (lane&15), na = 4*nstep + 2*(lane>>4)
//   PX : packed spectrum fragments   [m(361)][kstep(91)][lane(32)][8]
//        lane slot: (re[kk][c], re[kk+1][c], re[kk][c+16], re[kk+1][c+16],
//                    im ... same 4) with c = lane&15, kk = 4*kstep+2*(lane>>4)
// ---------------------------------------------------------------------------
#define PF_FLOATS ((size_t)NTILEM * NSTEP1 * 32 * 4)          // 529,920
#define PX_FLOATS ((size_t)MMAX * KSTEP2 * 32 * 8)            // 8,409,856

// ---------------------------------------------------------------------------
// Phase 0a: build packed DFT basis fragments (forward-norm rfft * 2*pi).
// Integer phase reduction keeps fp32 trig accurate at large n*m.
// ---------------------------------------------------------------------------
__global__ void sht_build_dft(float* __restrict__ PF) {
    int idx = blockIdx.x * blockDim.x + threadIdx.x;         // fragment id
    if (idx >= NTILEM * NSTEP1 * 32) return;
    int lane  = idx & 31;
    int nstep = (idx >> 5) % NSTEP1;
    int mtile = (idx >> 5) / NSTEP1;
    int li = lane & 15, half = lane >> 4;
    int m  = mtile * 16 + li;
    int na = nstep * 4 + 2 * half;

    v4f f = {0.0f, 0.0f, 0.0f, 0.0f};
    if (m < MMAX) {
        const float scale = TWO_PI / (float)NLON;
        int p0 = (na * m) % NLON;
        int p1 = ((na + 1) * m) % NLON;
        float t0 = (float)p0 * (TWO_PI / (float)NLON);
        float t1 = (float)p1 * (TWO_PI / (float)NLON);
        f.x = cosf(t0) * scale;
        f.y = cosf(t1) * scale;
        f.z = -sinf(t0) * scale;
        f.w = -sinf(t1) * scale;
    }
    ((v4f*)PF)[idx] = f;
}

// ---------------------------------------------------------------------------
// Phase 0b: zero the last k-step fragments of PX (k = 361..363 padding).
// ---------------------------------------------------------------------------
__global__ void sht_zero_tail(float* __restrict__ PX) {
    int idx = blockIdx.x * blockDim.x + threadIdx.x;
    if (idx >= MMAX * 256) return;                            // 32 lanes * 8
    int m = idx >> 8;
    int j = idx & 255;
    PX[((size_t)m * KSTEP2 + (KSTEP2 - 1)) * 256 + j] = 0.0f;
}

// ---------------------------------------------------------------------------
// Phase 1: DFT GEMM.  A = x as (11552 x 720), B = packed basis fragments.
// One wave per 16x16 tile; cos & sin share the A tile.  D scattered into PX
// in stage-2 fragment order.
// ---------------------------------------------------------------------------
__global__ __launch_bounds__(32) void sht_dft_stage1(
    const float* __restrict__ x, const float* __restrict__ PF,
    float* __restrict__ PX) {
    const int lane = threadIdx.x;
    const int half = lane >> 4;
    const int li   = lane & 15;

    // A row for this lane (rows are exact: 722*16 == 32*361)
    const int row = blockIdx.x * 16 + li;
    const int c   = row / NLAT;
    const int k   = row - c * NLAT;
    const v2f* __restrict__ xv =
        (const v2f*)(x + (size_t)(c * NLAT + k) * NLON);      // 8B-aligned

    const v4f* __restrict__ pf =
        (const v4f*)PF + (size_t)blockIdx.y * NSTEP1 * 32 + lane;

    v8f accC = {};
    v8f accS = {};

    for (int s = 0; s < NSTEP1; ++s) {
        v2f a = xv[s * 2 + half];                 // global_load_b64
        v4f f = pf[s * 32];                       // global_load_b128
        v2f bc = {f.x, f.y};
        v2f bs = {f.z, f.w};
        accC = __builtin_amdgcn_wmma_f32_16x16x4_f32(
            false, a, false, bc, (short)0, accC, false, false);
        accS = __builtin_amdgcn_wmma_f32_16x16x4_f32(
            false, a, false, bs, (short)0, accS, false, false);
    }

    const int mcol = blockIdx.y * 16 + li;        // D column = m
    if (mcol < MMAX) {
        // incremental row -> (c,k): one div for the tile base, then wrap
        const int rbase = blockIdx.x * 16;
        const int cb = rbase / NLAT;
        const int kb = rbase - cb * NLAT;
        const size_t mbase = (size_t)mcol * KSTEP2;
        #pragma unroll
        for (int v = 0; v < 8; ++v) {
            const int o = v + 8 * half;           // D row within tile
            int k2 = kb + o, c2 = cb;
            if (k2 >= NLAT) { k2 -= NLAT; c2 += 1; }
            const int lane2 = ((k2 & 2) << 3) + (c2 & 15);
            const int slot  = ((c2 >> 4) << 1) + (k2 & 1);
            float* frag = PX + ((mbase + (k2 >> 2)) * 32 + lane2) * 8;
            frag[slot]     = accC[v];             // re
            frag[slot + 4] = accS[v];             // im
        }
    }
}

// ---------------------------------------------------------------------------
// Phase 2: Legendre contraction.  Per m: out[c,l] = sum_k W[m,l,k]*X_m[k,c].
// A = W_m rows (clamped at l>=361; unused D rows masked at store),
// B = PX fragments (two b128 loads per lane per k-step).
// Main loop is guard-free; k-edge handled in a one-step epilogue.
// ---------------------------------------------------------------------------
__global__ __launch_bounds__(32) void sht_leg_stage2(
    const float* __restrict__ W, const float* __restrict__ PX,
    float* __restrict__ out) {
    const int m    = blockIdx.x;
    const int lane = threadIdx.x;
    const int half = lane >> 4;
    const int li   = lane & 15;

    const int l  = blockIdx.y * 16 + li;
    const int lc = (l < LMAX) ? l : (LMAX - 1);               // clamped row
    const float* __restrict__ wrow = W + ((size_t)m * LMAX + lc) * NLAT;

    // Prefetch this lane's weight row (361 floats = 1444 B) into cache.
    #pragma unroll
    for (int pfo = 0; pfo < 1444; pfo += 256)
        __builtin_prefetch((const char*)wrow + pfo, 0, 1);

    const v4f* __restrict__ pxp =
        (const v4f*)PX + ((size_t)m * KSTEP2 * 32 + lane) * 2;

    v8f aR0 = {}, aR1 = {}, aI0 = {}, aI1 = {};

    for (int s = 0; s < KSTEP2 - 1; ++s) {        // k <= 359: no guards
        const int kk = 4 * s + 2 * half;
        v2f a;
        a.x = wrow[kk];
        a.y = wrow[kk + 1];
        const v4f f0 = pxp[s * 64 + 0];           // re: c, c+16
        const v4f f1 = pxp[s * 64 + 1];           // im: c, c+16
        v2f br0 = {f0.x, f0.y};
        v2f br1 = {f0.z, f0.w};
        v2f bi0 = {f1.x, f1.y};
        v2f bi1 = {f1.z, f1.w};
        aR0 = __builtin_amdgcn_wmma_f32_16x16x4_f32(
            false, a, false, br0, (short)0, aR0, false, false);
        aR1 = __builtin_amdgcn_wmma_f32_16x16x4_f32(
            false, a, false, br1, (short)0, aR1, false, false);
        aI0 = __builtin_amdgcn_wmma_f32_16x16x4_f32(
            false, a, false, bi0, (short)0, aI0, false, false);
        aI1 = __builtin_amdgcn_wmma_f32_16x16x4_f32(
            false, a, false, bi1, (short)0, aI1, false, false);
    }
    {   // epilogue: k-step 90 covers k = 360 only (361..363 are zero-padded)
        const int s = KSTEP2 - 1;
        v2f a;
        a.x = (half == 0) ? wrow[360] : 0.0f;
        a.y = 0.0f;
        const v4f f0 = pxp[s * 64 + 0];
        const v4f f1 = pxp[s * 64 + 1];
        v2f br0 = {f0.x, f0.y};
        v2f br1 = {f0.z, f0.w};
        v2f bi0 = {f1.x, f1.y};
        v2f bi1 = {f1.z, f1.w};
        aR0 = __builtin_amdgcn_wmma_f32_16x16x4_f32(
            false, a, false, br0, (short)0, aR0, false, false);
        aR1 = __builtin_amdgcn_wmma_f32_16x16x4_f32(
            false, a, false, br1, (short)0, aR1, false, false);
        aI0 = __builtin_amdgcn_wmma_f32_16x16x4_f32(
            false, a, false, bi0, (short)0, aI0, false, false);
        aI1 = __builtin_amdgcn_wmma_f32_16x16x4_f32(
            false, a, false, bi1, (short)0, aI1, false, false);
    }

    // out layout: (1, 32, LMAX, MMAX, 2)
    #pragma unroll
    for (int v = 0; v < 8; ++v) {
        const int lrow = blockIdx.y * 16 + v + 8 * half;
        if (lrow < LMAX) {
            const int cA = li;
            const int cB = 16 + li;
            const size_t oA = (((size_t)cA * LMAX + lrow) * MMAX + m) * 2;
            const size_t oB = (((size_t)cB * LMAX + lrow) * MMAX + m) * 2;
            out[oA + 0] = aR0[v];
            out[oB + 0] = aR1[v];
            out[oA + 1] = aI0[v];
            out[oB + 1] = aI1[v];
        }
    }
}

extern "C" void kernel_launch(void* const* d_in, const int* in_sizes, int n_in,
                              void* d_out, int out_size, void* d_ws,
                              size_t ws_size, hipStream_t stream) {
    const float* x = (const float*)d_in[0];
    const float* w = (const float*)d_in[1];
    float* out = (float*)d_out;
    float* ws = (float*)d_ws;

    float* PF = ws;                     // 529,920 floats (~2.1 MB), 16B aligned
    float* PX = ws + PF_FLOATS;         // 8,409,856 floats (~33.6 MB)

    {   // Phase 0: packed basis + zero k-padding of PX
        int nf = NTILEM * NSTEP1 * 32;
        sht_build_dft<<<(nf + 255) / 256, 256, 0, stream>>>(PF);
        int nz = MMAX * 256;
        sht_zero_tail<<<(nz + 255) / 256, 256, 0, stream>>>(PX);
    }
    {   // Phase 1: DFT via f32 WMMA GEMM
        dim3 grid(NTILER, NTILEM);      // (722, 23)
        sht_dft_stage1<<<grid, 32, 0, stream>>>(x, PF, PX);
    }
    {   // Phase 2: Legendre contraction via f32 WMMA (batched over m)
        dim3 grid(MMAX, NTILEM);        // (361, 23)
        sht_leg_stage2<<<grid, 32, 0, stream>>>(w, PX, out);
    }
}